// DictionaryLearning_41369124995166
// MI455X (gfx1250) — compile-verified
//
#include <hip/hip_runtime.h>
#include <math.h>

typedef __attribute__((ext_vector_type(2))) float v2f;
typedef __attribute__((ext_vector_type(4))) float v4f;
typedef __attribute__((ext_vector_type(8))) float v8f;

#define NUM_EMB 512
#define EMB_DIM 64
#define KSP 8
#define EPSV 1e-10f
#define NB 32768           // 8*64*64 signals
#define HW 4096            // 64*64
#define ZE_N 2097152       // 8*64*4096
#define NBLK (NB / 16)     // 2048 OMP blocks

// workspace layout (float offsets)
#define WS_DN 0
#define WS_BOOST (WS_DN + EMB_DIM * NUM_EMB)     // 32768
#define WS_G (WS_BOOST + NUM_EMB)                // 33280
#define WS_PART (WS_G + NUM_EMB * NUM_EMB)       // 295424

// output layout (float offsets): z_dl | loss | support | coeffs
#define OUT_LOSS ZE_N
#define OUT_SUP (ZE_N + 1)
#define OUT_COEF (OUT_SUP + NB * KSP)

// ---------------- Kernel 1: normalize dictionary + selection boost ----------
__global__ __launch_bounds__(512) void prep_kernel(const float* __restrict__ dict,
                                                   const float* __restrict__ usage,
                                                   float* __restrict__ ws) {
  __shared__ float red[512];
  int n = threadIdx.x;                  // 0..511 : one atom per thread
  float* Dn = ws + WS_DN;
  float* boost = ws + WS_BOOST;
  float ss = 0.f;
  for (int c = 0; c < EMB_DIM; ++c) {
    float v = dict[c * NUM_EMB + n];
    ss += v * v;
  }
  float inv = 1.0f / fmaxf(sqrtf(ss), EPSV);
  for (int c = 0; c < EMB_DIM; ++c)
    Dn[c * NUM_EMB + n] = dict[c * NUM_EMB + n] * inv;

  float u = usage[n];
  red[n] = u;
  __syncthreads();
  for (int s = 256; s > 0; s >>= 1) {
    if (n < s) red[n] += red[n + s];
    __syncthreads();
  }
  float un = u / fmaxf(red[0], EPSV);
  float b = powf((1.0f / NUM_EMB) / fmaxf(un, EPSV), 0.3f);
  boost[n] = fminf(b, 8.0f);
}

// ---------------- Kernel 2: G = Dn^T Dn via f32 WMMA ------------------------
// One wave per 16x16 tile of G. M-dim = row atom ni, N-dim = col atom nj.
__global__ __launch_bounds__(32) void gram_kernel(float* __restrict__ ws) {
  const float* Dn = ws + WS_DN;
  float* G = ws + WS_G;
  int l = threadIdx.x;
  int half = l >> 4;          // 0: lanes 0-15, 1: lanes 16-31
  int lm = l & 15;
  int ni0 = blockIdx.x * 16;
  int nj0 = blockIdx.y * 16;
  v8f acc = {};
  for (int c = 0; c < EMB_DIM; c += 4) {
    int kb = c + 2 * half;    // A 16x4 f32 layout: VGPR0 K={0|2}, VGPR1 K={1|3}
    v2f a, b;
    a.x = Dn[(kb + 0) * NUM_EMB + ni0 + lm];
    a.y = Dn[(kb + 1) * NUM_EMB + ni0 + lm];
    b.x = Dn[(kb + 0) * NUM_EMB + nj0 + lm];
    b.y = Dn[(kb + 1) * NUM_EMB + nj0 + lm];
    acc = __builtin_amdgcn_wmma_f32_16x16x4_f32(false, a, false, b, (short)0,
                                                acc, false, false);
  }
  for (int r = 0; r < 8; ++r) {
    int m = r + 8 * half;     // C/D layout: rows r (lanes 0-15), r+8 (lanes 16-31)
    G[(ni0 + m) * NUM_EMB + nj0 + lm] = acc[r];
  }
}

// ---------------- Kernel 3: fused h_bar (WMMA) + OMP + outputs --------------
// Block = 512 threads = 16 waves; block owns 16 signals; one wave per signal.
__global__ __launch_bounds__(512) void omp_kernel(const float* __restrict__ z_e,
                                                  float* __restrict__ ws,
                                                  float* __restrict__ out) {
  __shared__ float hbar[16][520];        // 16 signals x 512 atoms (+16B-aligned pad)
  __shared__ float boost_s[512];
  __shared__ float gk_s[16][KSP];
  __shared__ int I_s[16][KSP];
  __shared__ float Ls[16][36];           // packed lower-tri 8x8
  __shared__ unsigned mask_s[16][16];    // 512-bit live mask per signal
  __shared__ float wsum[16];

  const float* Dn = ws + WS_DN;
  const float* G = ws + WS_G;
  int tid = threadIdx.x;
  int wv = tid >> 5;
  int lane = tid & 31;
  int s0 = blockIdx.x * 16;

  boost_s[tid] = ws[WS_BOOST + tid];
  if (lane < 16) mask_s[wv][lane] = 0xFFFFFFFFu;

  // ---- Phase A: hbar tile = X_tile^T * Dn via f32 WMMA (M = signal, N = atom)
  {
    int half = lane >> 4;
    int lm = lane & 15;
    int s = s0 + lm;
    int bb = s >> 12;                    // batch index (s / 4096)
    int hw = s & 4095;
    const float* Xb = z_e + (size_t)bb * EMB_DIM * HW + hw;  // X(c)=Xb[c*HW]
    for (int t = wv; t < 32; t += 16) {  // each wave does 2 atom-tiles
      int n0 = t * 16;
      v8f acc = {};
      for (int c = 0; c < EMB_DIM; c += 4) {
        int kb = c + 2 * half;
        v2f a, b;
        a.x = Xb[(size_t)(kb + 0) * HW];
        a.y = Xb[(size_t)(kb + 1) * HW];
        b.x = Dn[(kb + 0) * NUM_EMB + n0 + lm];
        b.y = Dn[(kb + 1) * NUM_EMB + n0 + lm];
        acc = __builtin_amdgcn_wmma_f32_16x16x4_f32(false, a, false, b, (short)0,
                                                    acc, false, false);
      }
      for (int r = 0; r < 8; ++r) hbar[r + 8 * half][n0 + lm] = acc[r];
    }
  }
  __syncthreads();

  // ---- Phase B: K OMP iterations, one wave per signal ----------------------
  const v4f* hb4 = (const v4f*)&hbar[wv][0];
  const v4f* boost4 = (const v4f*)&boost_s[0];
  for (int k = 1; k <= KSP; ++k) {
    // score = |hbar - sum_j gk_j * G[I_j,:]| * boost * mask
    // Each lane owns 4 consecutive atoms -> b128 loads from L2-resident G.
    float bestv = -1.0f;
    int bestn = 0;
    for (int c4 = 0; c4 < 4; ++c4) {
      int n0 = 128 * c4 + 4 * lane;      // 16B-aligned, coalesced across wave
      v4f h4 = hb4[32 * c4 + lane];
      for (int j = 0; j < k - 1; ++j) {
        const v4f* grow = (const v4f*)(G + (size_t)I_s[wv][j] * NUM_EMB);
        h4 -= gk_s[wv][j] * grow[32 * c4 + lane];
      }
      v4f b4 = boost4[32 * c4 + lane];
      unsigned mword = mask_s[wv][4 * c4 + (lane >> 3)] >> (n0 & 31);
      for (int e = 0; e < 4; ++e) {
        float sc = fabsf(h4[e]) * b4[e];
        sc = ((mword >> e) & 1u) ? sc : -1.0f;
        if (sc > bestv) { bestv = sc; bestn = n0 + e; }
      }
    }
    // wave32 butterfly argmax (max value, min index on tie -> all lanes agree)
    for (int off = 16; off > 0; off >>= 1) {
      float ov = __shfl_xor(bestv, off);
      int oi = __shfl_xor(bestn, off);
      if (ov > bestv || (ov == bestv && oi < bestn)) { bestv = ov; bestn = oi; }
    }
    // prefetch the freshly-selected G row (used next iteration) while lane 0
    // does the serial update: 32 lanes x 64B span = full 2KB row
    __builtin_prefetch(G + (size_t)bestn * NUM_EMB + lane * 16, 0, 1);
    if (lane == 0) {
      int idx = bestn;
      mask_s[wv][idx >> 5] &= ~(1u << (idx & 31));
      if (k == 1) {
        Ls[wv][0] = 1.0f;
      } else {                            // Cholesky rank-1 append
        float w[KSP];
        for (int j = 0; j < k - 1; ++j) {
          float rhs = G[(size_t)I_s[wv][j] * NUM_EMB + idx];
          for (int p = 0; p < j; ++p) rhs -= Ls[wv][j * (j + 1) / 2 + p] * w[p];
          w[j] = rhs / Ls[wv][j * (j + 1) / 2 + j];
        }
        float sw = 0.f;
        for (int j = 0; j < k - 1; ++j) sw += w[j] * w[j];
        float corner = sqrtf(fmaxf(1.0f - sw, 1e-12f));
        int row = k - 1;
        for (int p = 0; p < row; ++p) Ls[wv][row * (row + 1) / 2 + p] = w[p];
        Ls[wv][row * (row + 1) / 2 + row] = corner;
      }
      I_s[wv][k - 1] = idx;
      float y[KSP], g[KSP];
      for (int j = 0; j < k; ++j) {       // L y = h_I
        float rhs = hbar[wv][I_s[wv][j]];
        for (int p = 0; p < j; ++p) rhs -= Ls[wv][j * (j + 1) / 2 + p] * y[p];
        y[j] = rhs / Ls[wv][j * (j + 1) / 2 + j];
      }
      for (int j = k - 1; j >= 0; --j) {  // L^T g = y
        float rhs = y[j];
        for (int p = j + 1; p < k; ++p) rhs -= Ls[wv][p * (p + 1) / 2 + j] * g[p];
        g[j] = rhs / Ls[wv][j * (j + 1) / 2 + j];
      }
      for (int j = 0; j < k; ++j) gk_s[wv][j] = g[j];
    }
    __syncthreads();
  }

  // ---- Phase C: reconstruct, z_dl, loss partials, support/coeffs -----------
  {
    int s = s0 + wv;
    int bb = s >> 12;
    int hw = s & 4095;
    float err = 0.f;
    for (int c = lane; c < EMB_DIM; c += 32) {
      float zq = 0.f;
      for (int j = 0; j < KSP; ++j)
        zq += gk_s[wv][j] * Dn[c * NUM_EMB + I_s[wv][j]];
      size_t gi = (size_t)(bb * EMB_DIM + c) * HW + hw;
      float ze = z_e[gi];
      out[gi] = ze + (zq - ze);          // z_dl == z_q numerically
      float d = zq - ze;
      err += d * d;
    }
    for (int off = 16; off > 0; off >>= 1) err += __shfl_down(err, off);
    if (lane == 0) wsum[wv] = err;
  }
  __syncthreads();
  if (tid == 0) {
    float t = 0.f;
    for (int i = 0; i < 16; ++i) t += wsum[i];
    ws[WS_PART + blockIdx.x] = t;
  }
  if (lane < KSP) {
    int s = s0 + wv;
    out[OUT_SUP + (size_t)s * KSP + lane] = (float)I_s[wv][lane];
    out[OUT_COEF + (size_t)s * KSP + lane] = gk_s[wv][lane];
  }
}

// ---------------- Kernel 4: deterministic loss reduction --------------------
__global__ __launch_bounds__(256) void loss_kernel(const float* __restrict__ ws,
                                                   float* __restrict__ out) {
  __shared__ float red[256];
  int t = threadIdx.x;
  float s = 0.f;
  for (int i = t; i < NBLK; i += 256) s += ws[WS_PART + i];
  red[t] = s;
  __syncthreads();
  for (int o = 128; o > 0; o >>= 1) {
    if (t < o) red[t] += red[t + o];
    __syncthreads();
  }
  // loss = q_loss + 0.25*e_loss, both equal MSE numerically
  if (t == 0) out[OUT_LOSS] = 1.25f * red[0] / (float)ZE_N;
}

extern "C" void kernel_launch(void* const* d_in, const int* in_sizes, int n_in,
                              void* d_out, int out_size, void* d_ws, size_t ws_size,
                              hipStream_t stream) {
  (void)in_sizes; (void)n_in; (void)out_size; (void)ws_size;
  const float* z_e = (const float*)d_in[0];
  const float* dict = (const float*)d_in[1];
  const float* usage = (const float*)d_in[2];
  float* out = (float*)d_out;
  float* ws = (float*)d_ws;

  prep_kernel<<<1, 512, 0, stream>>>(dict, usage, ws);
  gram_kernel<<<dim3(32, 32), 32, 0, stream>>>(ws);
  omp_kernel<<<NBLK, 512, 0, stream>>>(z_e, ws, out);
  loss_kernel<<<1, 256, 0, stream>>>(ws, out);
}